// MaterialEmbeddingLayer_74208444940405
// MI455X (gfx1250) — compile-verified
//
#include <hip/hip_runtime.h>

// ---------------------------------------------------------------------------
// Problem constants (from reference)
// ---------------------------------------------------------------------------
#define NM   50000
#define NO   100000
#define NE   512000
#define DM   128
#define DOPR 192
#define DEA  64
#define DE   128

typedef __attribute__((ext_vector_type(2))) float v2f;
typedef __attribute__((ext_vector_type(8))) float v8f;

// fp32 WMMA: D(16x16) += A(16x4) * B(4x16), full fp32 precision
__device__ __forceinline__ v8f wmma_k4(v8f c, v2f a, v2f b) {
    return __builtin_amdgcn_wmma_f32_16x16x4_f32(
        /*neg_a=*/false, a, /*neg_b=*/false, b,
        /*c_mod=*/(short)0, c, /*reuse_a=*/false, /*reuse_b=*/false);
}

// ---------------------------------------------------------------------------
// Tensor Data Mover: DMA a 2D fp32 tile (rows x rowDwords) from global into
// LDS, with hardware padding (pad_amount dwords every 2^(padIntCode+1) dwords)
// so LDS row strides avoid bank conflicts. Issued by one wave per block;
// completion via s_wait_tensorcnt then the caller's workgroup barrier.
// 6-arg builtin flavor (clang-23 / therock-10.0 headers).
// ---------------------------------------------------------------------------
typedef __attribute__((ext_vector_type(4))) unsigned int tdm_g0_t;
typedef __attribute__((ext_vector_type(8))) int          tdm_g1_t;
typedef __attribute__((ext_vector_type(4))) int          tdm_g2_t;
typedef __attribute__((ext_vector_type(8))) int          tdm_g4_t;

__device__ __forceinline__ void tdm_load_2d(unsigned lds_byte_off,
                                            const void* gsrc,
                                            unsigned rowDwords, unsigned rows,
                                            unsigned padIntCode, unsigned padAmtM1)
{
    unsigned long long ga = (unsigned long long)(uintptr_t)gsrc;
    tdm_g0_t g0;
    g0[0] = 1u;                                           // count=1 (valid D#)
    g0[1] = lds_byte_off;                                 // lds_addr
    g0[2] = (unsigned)ga;                                 // global_addr[31:0]
    g0[3] = (unsigned)((ga >> 32) & 0x1FFFFFFu) | (2u << 30); // addr[56:32], type=2
    tdm_g1_t g1;
    // data_size=2 (4B) | pad_enable | pad_interval | pad_amount
    g1[0] = (int)((2u << 16) | (1u << 20) | (padIntCode << 22) | (padAmtM1 << 25));
    g1[1] = (int)((rowDwords & 0xFFFFu) << 16);           // tensor_dim0[15:0]
    g1[2] = (int)(((rowDwords >> 16) & 0xFFFFu) | ((rows & 0xFFFFu) << 16)); // dim0 hi, dim1 lo
    g1[3] = (int)(((rows >> 16) & 0xFFFFu) | ((rowDwords & 0xFFFFu) << 16)); // dim1 hi, tile_dim0
    g1[4] = (int)(rows & 0xFFFFu);                        // tile_dim1 (tile_dim2=0)
    g1[5] = (int)rowDwords;                               // tensor_dim0_stride lo
    g1[6] = 0;
    g1[7] = 0;
    tdm_g2_t z4 = {0, 0, 0, 0};
    tdm_g4_t z8 = {0, 0, 0, 0, 0, 0, 0, 0};
    __builtin_amdgcn_tensor_load_to_lds(g0, g1, z4, z4, z8, 0); // tensor_load_to_lds
    __builtin_amdgcn_s_wait_tensorcnt(0);                       // s_wait_tensorcnt 0
}

// ---------------------------------------------------------------------------
// K0: zero a float buffer
// ---------------------------------------------------------------------------
__global__ __launch_bounds__(256) void k_zero(float* __restrict__ p, int n) {
    int i = blockIdx.x * 256 + threadIdx.x;
    if (i < n) p[i] = 0.f;
}

// ---------------------------------------------------------------------------
// K1: m = materials @ W_mat  (+ fused self-attention score)
// 1 block = 1 M-tile (16 rows, TDM-staged in LDS); 8 waves = 8 N-tiles; K=128
// LDS row stride 132 dwords (TDM pad 4/128) -> conflict-free ds_load_b64
// ---------------------------------------------------------------------------
__global__ __launch_bounds__(256) void k_gemm_mat(
    const float* __restrict__ A,     // NM x 128
    const float* __restrict__ W,     // 128 x 128
    const float* __restrict__ attS,  // 256
    float* __restrict__ M,           // NM x 128
    float* __restrict__ sSelf)       // NM
{
    __shared__ float sA[16 * 132];
    __shared__ float sdot[16];
    const int t    = threadIdx.x;
    const int wave = t >> 5;
    const int lane = t & 31;
    const int half = lane >> 4;
    const int l16  = lane & 15;
    const int row0 = blockIdx.x * 16;
    const int col0 = wave * 16;

    if (t == 0)
        tdm_load_2d((unsigned)(uintptr_t)sA, A + (size_t)row0 * DM,
                    DM, 16, /*128 dw*/6, /*4 dw*/3);
    if (t < 16) sdot[t] = 0.f;
    __syncthreads();

    v8f acc = {};
    const float* arow = sA + l16 * 132;
    #pragma unroll 4
    for (int k = 0; k < DM; k += 4) {
        const int ka = k + 2 * half;
        v2f a; a[0] = arow[ka];                           // ds_load_b64
               a[1] = arow[ka + 1];
        v2f b; b[0] = W[(size_t)ka * DE + col0 + l16];
               b[1] = W[(size_t)(ka + 1) * DE + col0 + l16];
        acc = wmma_k4(acc, a, b);
    }

    const int   col = col0 + l16;
    const float ac  = attS[col] + attS[DE + col];   // concat([m,m])@att_self
    #pragma unroll
    for (int j = 0; j < 8; ++j) {
        const int r = j + 8 * half;
        const float v = acc[j];
        M[(size_t)(row0 + r) * DE + col] = v;
        atomicAdd(&sdot[r], v * ac);                // ds_add_f32
    }
    __syncthreads();
    if (t < 16) {
        const float s = sdot[t];
        sSelf[row0 + t] = (s > 0.f) ? s : 0.2f * s; // leaky-relu(0.2)
    }
}

// ---------------------------------------------------------------------------
// K2: P = operations @ W_op[0:192,:]   (hoisted out of the edge loop)
// A-tile 16x192 TDM-staged; padded element (r,c) lives at d + 4*(d>>7),
// d = r*192+c  (pairs never straddle a pad since d is even)
// ---------------------------------------------------------------------------
__global__ __launch_bounds__(256) void k_gemm_op(
    const float* __restrict__ A,     // NO x 192
    const float* __restrict__ W,     // rows 0..191 of W_op (ld=128)
    float* __restrict__ P)           // NO x 128
{
    __shared__ float sA[3168];       // 3072 data dwords + 24 pads * 4
    const int t    = threadIdx.x;
    const int wave = t >> 5;
    const int lane = t & 31;
    const int half = lane >> 4;
    const int l16  = lane & 15;
    const int row0 = blockIdx.x * 16;
    const int col0 = wave * 16;

    if (t == 0)
        tdm_load_2d((unsigned)(uintptr_t)sA, A + (size_t)row0 * DOPR,
                    DOPR, 16, /*128 dw*/6, /*4 dw*/3);
    __syncthreads();

    v8f acc = {};
    #pragma unroll 4
    for (int k = 0; k < DOPR; k += 4) {
        const int ka = k + 2 * half;
        const int d  = l16 * DOPR + ka;
        const int o  = d + ((d >> 7) << 2);               // pad-adjusted offset
        v2f a; a[0] = sA[o];
               a[1] = sA[o + 1];
        v2f b; b[0] = W[(size_t)ka * DE + col0 + l16];
               b[1] = W[(size_t)(ka + 1) * DE + col0 + l16];
        acc = wmma_k4(acc, a, b);
    }
    const int col = col0 + l16;
    #pragma unroll
    for (int j = 0; j < 8; ++j)
        P[(size_t)(row0 + j + 8 * half) * DE + col] = acc[j];
}

// ---------------------------------------------------------------------------
// Shared edge-tile computation: wave's 16x16 sub-tile of
// ops_e = edge_attr@W_op_bot + P[src], with edge_attr tile already in LDS
// (row stride 68 dwords via TDM pad -> conflict-free)
// ---------------------------------------------------------------------------
__device__ __forceinline__ v8f edge_tile(
    const float* __restrict__ sA, const float* __restrict__ Wb,
    const float* __restrict__ P,  const int* __restrict__ s_src,
    int wave, int lane)
{
    const int half = lane >> 4;
    const int l16  = lane & 15;
    const int col0 = wave * 16;

    v8f acc = {};
    const float* arow = sA + l16 * 68;
    #pragma unroll 4
    for (int k = 0; k < DEA; k += 4) {
        const int ka = k + 2 * half;
        v2f a; a[0] = arow[ka];
               a[1] = arow[ka + 1];
        v2f b; b[0] = Wb[(size_t)ka * DE + col0 + l16];
               b[1] = Wb[(size_t)(ka + 1) * DE + col0 + l16];
        acc = wmma_k4(acc, a, b);
    }
    const int col = col0 + l16;
    #pragma unroll
    for (int j = 0; j < 8; ++j)
        acc[j] += P[(size_t)s_src[j + 8 * half] * DE + col]; // gather add
    return acc;
}

// ---------------------------------------------------------------------------
// K3: cross-attention scores per edge (ops_e never materialized)
// ---------------------------------------------------------------------------
__global__ __launch_bounds__(256) void k_edge_score(
    const float* __restrict__ EA,    // E x 64
    const int*   __restrict__ esrc,
    const int*   __restrict__ edst,
    const float* __restrict__ Wb,    // W_op rows 192..255 (64 x 128)
    const float* __restrict__ P,     // NO x 128
    const float* __restrict__ M,     // NM x 128
    const float* __restrict__ attC,  // 256
    float* __restrict__ sCross)      // E
{
    __shared__ float sA[16 * 68];
    __shared__ int   s_src[16];
    __shared__ int   s_dst[16];
    __shared__ float s_sc[16];
    const int e0 = blockIdx.x * 16;
    const int t  = threadIdx.x;
    if (t == 0)
        tdm_load_2d((unsigned)(uintptr_t)sA, EA + (size_t)e0 * DEA,
                    DEA, 16, /*64 dw*/5, /*4 dw*/3);
    if (t < 16) {
        s_src[t] = esrc[e0 + t];
        s_dst[t] = edst[e0 + t];
        s_sc[t]  = 0.f;
    }
    __syncthreads();

    // mat_e . att_cross[0:128] : 16 threads per edge row, 8 cols each
    {
        const int r = t >> 4, cb = (t & 15) * 8;
        const float* mrow = M + (size_t)s_dst[r] * DE;
        float p = 0.f;
        #pragma unroll
        for (int c = 0; c < 8; ++c) p += mrow[cb + c] * attC[cb + c];
        atomicAdd(&s_sc[r], p);
    }

    // ops_e . att_cross[128:256] via WMMA tile
    const int wave = t >> 5, lane = t & 31;
    v8f acc = edge_tile(sA, Wb, P, s_src, wave, lane);
    const int   col = wave * 16 + (lane & 15);
    const float ac2 = attC[DE + col];
    #pragma unroll
    for (int j = 0; j < 8; ++j)
        atomicAdd(&s_sc[j + 8 * (lane >> 4)], acc[j] * ac2);

    __syncthreads();
    if (t < 16) {
        const float s = s_sc[t];
        sCross[e0 + t] = (s > 0.f) ? s : 0.2f * s;
    }
}

// ---------------------------------------------------------------------------
// K4: deterministic tree reductions for global softmax (max, then sum(exp))
// ---------------------------------------------------------------------------
#define RB 1024
__global__ __launch_bounds__(256) void k_partial_max(
    const float* __restrict__ s1, int n1,
    const float* __restrict__ s2, int n2,
    float* __restrict__ pmax)
{
    __shared__ float red[256];
    const int total = n1 + n2;
    float mx = -3.402823466e38f;
    for (int i = blockIdx.x * 256 + threadIdx.x; i < total; i += gridDim.x * 256)
        mx = fmaxf(mx, (i < n1) ? s1[i] : s2[i - n1]);
    red[threadIdx.x] = mx; __syncthreads();
    for (int s = 128; s > 0; s >>= 1) {
        if (threadIdx.x < s) red[threadIdx.x] = fmaxf(red[threadIdx.x], red[threadIdx.x + s]);
        __syncthreads();
    }
    if (threadIdx.x == 0) pmax[blockIdx.x] = red[0];
}

__global__ __launch_bounds__(256) void k_final_max(
    const float* __restrict__ pmax, float* __restrict__ stats)
{
    __shared__ float red[256];
    float mx = -3.402823466e38f;
    for (int i = threadIdx.x; i < RB; i += 256) mx = fmaxf(mx, pmax[i]);
    red[threadIdx.x] = mx; __syncthreads();
    for (int s = 128; s > 0; s >>= 1) {
        if (threadIdx.x < s) red[threadIdx.x] = fmaxf(red[threadIdx.x], red[threadIdx.x + s]);
        __syncthreads();
    }
    if (threadIdx.x == 0) stats[0] = red[0];
}

__global__ __launch_bounds__(256) void k_partial_sum(
    const float* __restrict__ s1, int n1,
    const float* __restrict__ s2, int n2,
    const float* __restrict__ stats, float* __restrict__ psum)
{
    __shared__ float red[256];
    const float gmax = stats[0];
    const int total = n1 + n2;
    float acc = 0.f;
    for (int i = blockIdx.x * 256 + threadIdx.x; i < total; i += gridDim.x * 256)
        acc += __expf(((i < n1) ? s1[i] : s2[i - n1]) - gmax);
    red[threadIdx.x] = acc; __syncthreads();
    for (int s = 128; s > 0; s >>= 1) {
        if (threadIdx.x < s) red[threadIdx.x] += red[threadIdx.x + s];
        __syncthreads();
    }
    if (threadIdx.x == 0) psum[blockIdx.x] = red[0];
}

__global__ __launch_bounds__(256) void k_final_sum(
    const float* __restrict__ psum, float* __restrict__ stats)
{
    __shared__ float red[256];
    float acc = 0.f;
    for (int i = threadIdx.x; i < RB; i += 256) acc += psum[i];
    red[threadIdx.x] = acc; __syncthreads();
    for (int s = 128; s > 0; s >>= 1) {
        if (threadIdx.x < s) red[threadIdx.x] += red[threadIdx.x + s];
        __syncthreads();
    }
    if (threadIdx.x == 0) stats[1] = red[0];
}

// ---------------------------------------------------------------------------
// K5: recompute ops_e tile, scatter norm_cross * ops_e into agg (f32 atomics)
// ---------------------------------------------------------------------------
__global__ __launch_bounds__(256) void k_edge_scatter(
    const float* __restrict__ EA,
    const int*   __restrict__ esrc,
    const int*   __restrict__ edst,
    const float* __restrict__ Wb,
    const float* __restrict__ P,
    const float* __restrict__ sCross,
    const float* __restrict__ stats,
    float* __restrict__ agg)         // NM x 128
{
    __shared__ float sA[16 * 68];
    __shared__ int   s_src[16];
    __shared__ int   s_dst[16];
    __shared__ float s_w[16];
    const int e0 = blockIdx.x * 16;
    const int t  = threadIdx.x;
    if (t == 0)
        tdm_load_2d((unsigned)(uintptr_t)sA, EA + (size_t)e0 * DEA,
                    DEA, 16, /*64 dw*/5, /*4 dw*/3);
    if (t < 16) {
        s_src[t] = esrc[e0 + t];
        s_dst[t] = edst[e0 + t];
        s_w[t]   = __expf(sCross[e0 + t] - stats[0]) / stats[1];
    }
    __syncthreads();

    const int wave = t >> 5, lane = t & 31;
    v8f acc = edge_tile(sA, Wb, P, s_src, wave, lane);
    const int col = wave * 16 + (lane & 15);
    #pragma unroll
    for (int j = 0; j < 8; ++j) {
        const int r = j + 8 * (lane >> 4);
        atomicAdd(&agg[(size_t)s_dst[r] * DE + col], s_w[r] * acc[j]); // global_atomic_add_f32
    }
}

// ---------------------------------------------------------------------------
// K6: out = elu(norm_self * m + agg)
// ---------------------------------------------------------------------------
__global__ __launch_bounds__(256) void k_finalize(
    const float* __restrict__ M,
    const float* __restrict__ agg,
    const float* __restrict__ sSelf,
    const float* __restrict__ stats,
    float* __restrict__ out)
{
    const int i = blockIdx.x * 256 + threadIdx.x;
    if (i >= NM * DE) return;
    const int row = i >> 7;
    const float ns = __expf(sSelf[row] - stats[0]) / stats[1];
    const float x  = ns * M[i] + agg[i];
    out[i] = (x > 0.f) ? x : expm1f(x);
}

// ---------------------------------------------------------------------------
// Host launcher
// ---------------------------------------------------------------------------
extern "C" void kernel_launch(void* const* d_in, const int* in_sizes, int n_in,
                              void* d_out, int out_size, void* d_ws, size_t ws_size,
                              hipStream_t stream) {
    const float* materials = (const float*)d_in[0];
    const float* operations = (const float*)d_in[1];
    const float* edge_attr = (const float*)d_in[2];
    const int*   edge_src  = (const int*)d_in[3];
    const int*   edge_dst  = (const int*)d_in[4];
    const float* W_mat     = (const float*)d_in[5];
    const float* W_op      = (const float*)d_in[6];
    const float* att_self  = (const float*)d_in[7];
    const float* att_cross = (const float*)d_in[8];
    float* out = (float*)d_out;

    // workspace layout (floats)
    float* ws = (float*)d_ws;
    float* M      = ws;                                   // NM*128
    float* P      = M + (size_t)NM * DE;                  // NO*128
    float* agg    = P + (size_t)NO * DE;                  // NM*128
    float* sSelf  = agg + (size_t)NM * DE;                // NM
    float* sCross = sSelf + NM;                           // NE
    float* pmax   = sCross + NE;                          // RB
    float* psum   = pmax + RB;                            // RB
    float* stats  = psum + RB;                            // 2

    const float* W_op_top = W_op;                         // rows 0..191
    const float* W_op_bot = W_op + (size_t)DOPR * DE;     // rows 192..255

    // 0) zero the aggregation buffer (ws is poisoned, and replays accumulate)
    k_zero<<<(NM * DE + 255) / 256, 256, 0, stream>>>(agg, NM * DE);

    // 1) m = materials @ W_mat, fused self-attention scores
    k_gemm_mat<<<NM / 16, 256, 0, stream>>>(materials, W_mat, att_self, M, sSelf);

    // 2) P = operations @ W_op_top (hoisted per-operation GEMM)
    k_gemm_op<<<NO / 16, 256, 0, stream>>>(operations, W_op_top, P);

    // 3) cross-attention scores per edge
    k_edge_score<<<NE / 16, 256, 0, stream>>>(edge_attr, edge_src, edge_dst,
                                              W_op_bot, P, M, att_cross, sCross);

    // 4) global softmax stats (deterministic tree reductions)
    k_partial_max<<<RB, 256, 0, stream>>>(sSelf, NM, sCross, NE, pmax);
    k_final_max<<<1, 256, 0, stream>>>(pmax, stats);
    k_partial_sum<<<RB, 256, 0, stream>>>(sSelf, NM, sCross, NE, stats, psum);
    k_final_sum<<<1, 256, 0, stream>>>(psum, stats);

    // 5) scatter norm_cross * ops_e into agg
    k_edge_scatter<<<NE / 16, 256, 0, stream>>>(edge_attr, edge_src, edge_dst,
                                                W_op_bot, P, sCross, stats, agg);

    // 6) out = elu(norm_self * m + agg)
    k_finalize<<<(NM * DE + 255) / 256, 256, 0, stream>>>(M, agg, sSelf, stats, out);
}